// GraphConvolution_4664334483852
// MI455X (gfx1250) — compile-verified
//
#include <hip/hip_runtime.h>

typedef __attribute__((ext_vector_type(2))) float v2f;
typedef __attribute__((ext_vector_type(4))) float v4f;
typedef __attribute__((ext_vector_type(8))) float v8f;

#define N_NODES 50000
#define D_IN    128
#define D_OUT   128
#define N_EDGES 800000

// ---------------------------------------------------------------------------
// Kernel 0: out[i][j] = bias[j]   (fold bias into the scatter target)
// ---------------------------------------------------------------------------
__global__ void gcn_init_bias(const float* __restrict__ bias,
                              float* __restrict__ out) {
  int i = blockIdx.x * blockDim.x + threadIdx.x;
  if (i < N_NODES * D_OUT) out[i] = bias[i & (D_OUT - 1)];
}

// ---------------------------------------------------------------------------
// Kernel 1: support = x @ W   via V_WMMA_F32_16X16X4_F32 (fp32-exact path).
// One wave per block. Each wave computes a 16(M) x 64(N) strip:
//   4 accumulators, A-fragment reused across 4 WMMAs per K-step, 32 K-steps.
//
// A 16x4 f32 frag (2 VGPRs): lanes 0-15 -> M=lane, K = {0,1}; lanes 16-31 ->
// M=lane-16, K = {2,3}.  B 4x16 symmetric over N.  C/D 16x16: VGPR v holds
// row v (lanes 0-15) and row v+8 (lanes 16-31).
// ---------------------------------------------------------------------------
__global__ __launch_bounds__(32) void gcn_gemm_wmma(
    const float* __restrict__ x,
    const float* __restrict__ w,
    float* __restrict__ support) {
  const int tile = blockIdx.x;
  const int mt   = tile >> 1;     // 0..3124  (16-row tiles; 50000/16 = 3125)
  const int nt   = tile & 1;      // 0..1     (64-col halves of D_OUT=128)
  const int lane = threadIdx.x & 31;
  const int half = lane >> 4;     // 0: K={0,1} / rows 0-7; 1: K={2,3} / rows 8-15
  const int l15  = lane & 15;

  const int row     = mt * 16 + l15;   // A-matrix row this lane loads
  const int colbase = nt * 64 + l15;   // B-matrix column base this lane loads

  v8f acc0 = {}, acc1 = {}, acc2 = {}, acc3 = {};

  for (int k0 = 0; k0 < D_IN; k0 += 4) {
    const int ka = k0 + half * 2;

    v2f a;
    a.x = x[row * D_IN + ka];
    a.y = x[row * D_IN + ka + 1];

    const float* wr0 = w + ka * D_OUT;
    const float* wr1 = w + (ka + 1) * D_OUT;
    v2f b0, b1, b2, b3;
    b0.x = wr0[colbase +  0]; b0.y = wr1[colbase +  0];
    b1.x = wr0[colbase + 16]; b1.y = wr1[colbase + 16];
    b2.x = wr0[colbase + 32]; b2.y = wr1[colbase + 32];
    b3.x = wr0[colbase + 48]; b3.y = wr1[colbase + 48];

    acc0 = __builtin_amdgcn_wmma_f32_16x16x4_f32(false, a, false, b0,
                                                 (short)0, acc0, false, false);
    acc1 = __builtin_amdgcn_wmma_f32_16x16x4_f32(false, a, false, b1,
                                                 (short)0, acc1, false, false);
    acc2 = __builtin_amdgcn_wmma_f32_16x16x4_f32(false, a, false, b2,
                                                 (short)0, acc2, false, false);
    acc3 = __builtin_amdgcn_wmma_f32_16x16x4_f32(false, a, false, b3,
                                                 (short)0, acc3, false, false);
  }

  // Store D: VGPR v -> output row (mt*16 + half*8 + v), column colbase + n*16.
  const int orow = mt * 16 + half * 8;
  #pragma unroll
  for (int v = 0; v < 8; ++v) {
    float* o = support + (orow + v) * D_OUT + nt * 64 + l15;
    o[0]  = acc0[v];
    o[16] = acc1[v];
    o[32] = acc2[v];
    o[48] = acc3[v];
  }
}

// ---------------------------------------------------------------------------
// Kernel 2: edge scatter. One wave per edge: lanes share (row,col,val) via
// scalar-broadcast loads, each lane gathers a float4 of the support row
// (L2-resident: 25.6 MB << 192 MB) and does 4 global_atomic_add_f32.
// ---------------------------------------------------------------------------
__global__ __launch_bounds__(256) void gcn_spmm_scatter(
    const int* __restrict__ adj_row,
    const int* __restrict__ adj_col,
    const float* __restrict__ adj_val,
    const float* __restrict__ support,
    float* __restrict__ out) {
  const int gid  = blockIdx.x * blockDim.x + threadIdx.x;
  const int e    = gid >> 5;
  const int lane = gid & 31;
  if (e >= N_EDGES) return;

  const int   r = adj_row[e];
  const int   c = adj_col[e];
  const float v = adj_val[e];

  const v4f m = *((const v4f*)(support + c * D_OUT) + lane);
  float* o = out + r * D_OUT + lane * 4;
  atomicAdd(o + 0, v * m.x);
  atomicAdd(o + 1, v * m.y);
  atomicAdd(o + 2, v * m.z);
  atomicAdd(o + 3, v * m.w);
}

// ---------------------------------------------------------------------------
extern "C" void kernel_launch(void* const* d_in, const int* in_sizes, int n_in,
                              void* d_out, int out_size, void* d_ws, size_t ws_size,
                              hipStream_t stream) {
  const float* x       = (const float*)d_in[0];
  const float* weight  = (const float*)d_in[1];
  const float* bias    = (const float*)d_in[2];
  const int*   adj_row = (const int*)d_in[3];
  const int*   adj_col = (const int*)d_in[4];
  const float* adj_val = (const float*)d_in[5];
  float* out     = (float*)d_out;
  float* support = (float*)d_ws;   // 50000*128*4 = 25.6 MB scratch

  // 1) out = broadcast(bias)
  {
    int total = N_NODES * D_OUT;
    gcn_init_bias<<<(total + 255) / 256, 256, 0, stream>>>(bias, out);
  }

  // 2) support = x @ W  (WMMA f32, one wave per 16x64 strip)
  {
    int grid = (N_NODES / 16) * (D_OUT / 64);   // 3125 * 2 = 6250 waves
    gcn_gemm_wmma<<<grid, 32, 0, stream>>>(x, weight, support);
  }

  // 3) out[row] += val * support[col]  (wave per edge, f32 atomics into L2)
  {
    long long threads = (long long)N_EDGES * 32;
    int blocks = (int)((threads + 255) / 256);
    gcn_spmm_scatter<<<blocks, 256, 0, stream>>>(adj_row, adj_col, adj_val,
                                                 support, out);
  }
}